// MeshRasterizer_5222680231953
// MI455X (gfx1250) — compile-verified
//
#include <hip/hip_runtime.h>
#include <hip/hip_bf16.h>

// Problem constants (match reference)
#define B_ 2
#define V_ 8000
#define F_ 1500
#define H_ 128
#define W_ 128
#define EPSA 1e-08f
#define EPSZ 1e-04f

#define NGROUPS ((F_ + 15) / 16)      // 94
#define FPAD (NGROUPS * 16)           // 1504
#define NP (B_ * H_ * W_)             // 32768

typedef float v2f __attribute__((ext_vector_type(2)));
typedef float v8f __attribute__((ext_vector_type(8)));

// ---------------------------------------------------------------------------
// Kernel 1: project vertices.  vv = verts @ R + T ; xy = focal*xy/z
// ---------------------------------------------------------------------------
__global__ void mr_project(const float* __restrict__ verts,
                           const float* __restrict__ R,
                           const float* __restrict__ T,
                           const float* __restrict__ focal,
                           float* __restrict__ vs) {
  int i = blockIdx.x * blockDim.x + threadIdx.x;
  if (i >= B_ * V_) return;
  int b = i / V_;
  const float* vp = verts + (size_t)i * 3;
  const float* Rp = R + (size_t)b * 9;
  const float* Tp = T + (size_t)b * 3;
  float x = vp[0], y = vp[1], z = vp[2];
  // vv[j] = sum_i verts[i] * R[i][j] + T[j]   (einsum 'bvi,bij->bvj')
  float vx = x * Rp[0] + y * Rp[3] + z * Rp[6] + Tp[0];
  float vy = x * Rp[1] + y * Rp[4] + z * Rp[7] + Tp[1];
  float vz = x * Rp[2] + y * Rp[5] + z * Rp[8] + Tp[2];
  float f = focal[0];
  vs[(size_t)i * 3 + 0] = f * vx / vz;
  vs[(size_t)i * 3 + 1] = f * vy / vz;
  vs[(size_t)i * 3 + 2] = vz;
}

// ---------------------------------------------------------------------------
// Kernel 2: per-face affine coefficients, packed in WMMA A-matrix lane layout.
// Channel ch in {w0,w1,w2,zp}; w_i and zp are affine in (px,py); all coeffs
// pre-divided by area_safe so WMMA directly yields barycentrics & zp.
// A-matrix 16x4 layout: lanes 0-15 hold (K0,K1)=(cpx,cpy) of face m=lane;
// lanes 16-31 hold (K2,K3)=(cconst, 0) of face m=lane-16.
// apack layout: [b][group][ch][lane] float2.
// ---------------------------------------------------------------------------
__global__ void mr_facecoef(const float* __restrict__ vs,
                            const int* __restrict__ faces,
                            float* __restrict__ fv,
                            float2* __restrict__ apack) {
  int i = blockIdx.x * blockDim.x + threadIdx.x;
  if (i >= B_ * FPAD) return;
  int b = i / FPAD;
  int f = i % FPAD;

  float c[4][3];
#pragma unroll
  for (int ch = 0; ch < 4; ++ch) { c[ch][0] = 0.f; c[ch][1] = 0.f; c[ch][2] = 0.f; }
  c[0][2] = -1e30f;  // default: invalid (w0 always negative -> never inside)

  if (f < F_) {
    int i0 = faces[f * 3 + 0], i1 = faces[f * 3 + 1], i2 = faces[f * 3 + 2];
    const float* p0 = vs + ((size_t)b * V_ + i0) * 3;
    const float* p1 = vs + ((size_t)b * V_ + i1) * 3;
    const float* p2 = vs + ((size_t)b * V_ + i2) * 3;
    float x0 = p0[0], y0 = p0[1], z0 = p0[2];
    float x1 = p1[0], y1 = p1[1], z1 = p1[2];
    float x2 = p2[0], y2 = p2[1], z2 = p2[2];

    // stash verts for the epilogue
    float* fp = fv + ((size_t)b * F_ + f) * 9;
    fp[0] = x0; fp[1] = y0; fp[2] = z0;
    fp[3] = x1; fp[4] = y1; fp[5] = z1;
    fp[6] = x2; fp[7] = y2; fp[8] = z2;

    // edge(a,b,p) = px*(by-ay) + py*(ax-bx) + (ay*bx - ax*by)
    float a0x = y2 - y1, a0y = x1 - x2, a0c = y1 * x2 - x1 * y2;  // w0: (v1,v2)
    float a1x = y0 - y2, a1y = x2 - x0, a1c = y2 * x0 - x2 * y0;  // w1: (v2,v0)
    float a2x = y1 - y0, a2y = x0 - x1, a2c = y0 * x1 - x0 * y1;  // w2: (v0,v1)
    float area = (x2 - x0) * (y1 - y0) - (y2 - y0) * (x1 - x0);
    if (fabsf(area) > EPSA) {
      float inv = 1.0f / area;  // area_safe == area here
      c[0][0] = a0x * inv; c[0][1] = a0y * inv; c[0][2] = a0c * inv;
      c[1][0] = a1x * inv; c[1][1] = a1y * inv; c[1][2] = a1c * inv;
      c[2][0] = a2x * inv; c[2][1] = a2y * inv; c[2][2] = a2c * inv;
      c[3][0] = (a0x * z0 + a1x * z1 + a2x * z2) * inv;
      c[3][1] = (a0y * z0 + a1y * z1 + a2y * z2) * inv;
      c[3][2] = (a0c * z0 + a1c * z1 + a2c * z2) * inv;
    }
    // |area| <= EPS -> keep invalid default (reference: inside requires |area|>EPS)
  }

  int gid = f >> 4, m = f & 15;
  float2* base = apack + (((size_t)b * NGROUPS + gid) * 4) * 32;
#pragma unroll
  for (int ch = 0; ch < 4; ++ch) {
    base[ch * 32 + m]      = make_float2(c[ch][0], c[ch][1]);  // K0,K1 lane
    base[ch * 32 + m + 16] = make_float2(c[ch][2], 0.0f);      // K2,K3 lane
  }
}

// ---------------------------------------------------------------------------
// Kernel 3: rasterize. One workgroup per (b,row); 8 waves x 16 pixels = W.
// Per 16-face group: 4x V_WMMA_F32_16X16X4_F32 produce (b0,b1,b2,zp) for
// 16 faces x 16 pixels. Lane L holds pixel N=L%16, faces M=j+8*(L/16).
// ---------------------------------------------------------------------------
__device__ __forceinline__ float mr_edge(float ax, float ay, float bx, float by,
                                         float px, float py) {
  return (px - ax) * (by - ay) - (py - ay) * (bx - ax);
}

__device__ __forceinline__ float mr_seg_d2(float ax, float ay, float bx, float by,
                                           float px, float py) {
  float dx = bx - ax, dy = by - ay;
  float l2 = fmaxf(dx * dx + dy * dy, 1e-12f);
  float t = ((px - ax) * dx + (py - ay) * dy) / l2;
  t = fminf(fmaxf(t, 0.0f), 1.0f);
  float ex = ax + t * dx - px, ey = ay + t * dy - py;
  return ex * ex + ey * ey;
}

__global__ void __launch_bounds__(256)
mr_raster(const float* __restrict__ apackf,
          const float* __restrict__ fv,
          float* __restrict__ out) {
  const int lane = threadIdx.x & 31;
  const int wave = threadIdx.x >> 5;
  const int b = blockIdx.x / H_;
  const int row = blockIdx.x % H_;
  const int n = lane & 15;
  const int hi = lane >> 4;
  const int col = wave * 16 + n;

  const float px = 1.0f - 2.0f * ((float)col + 0.5f) / (float)W_;
  const float py = 1.0f - 2.0f * ((float)row + 0.5f) / (float)H_;

  // B-matrix 4x16: VGPR0 = rows K0/K2 (lanes 0-15 / 16-31), VGPR1 = K1/K3.
  v2f bm;
  bm.x = (lane < 16) ? px : 1.0f;   // B[0][n]=px ; B[2][n]=1
  bm.y = (lane < 16) ? py : 0.0f;   // B[1][n]=py ; B[3][n]=0

  v8f zero = {0.f, 0.f, 0.f, 0.f, 0.f, 0.f, 0.f, 0.f};

  float zmin = __builtin_inff();
  int idxm = 0x7fffffff;

  const v2f* ap = (const v2f*)apackf;
  const size_t bbase = (size_t)b * NGROUPS * 4 * 32;

  for (int g = 0; g < NGROUPS; ++g) {
    const v2f* gp = ap + bbase + (size_t)g * 4 * 32;
    __builtin_prefetch((const void*)(gp + 4 * 32), 0, 3);

    v2f a0 = gp[0 * 32 + lane];
    v2f a1 = gp[1 * 32 + lane];
    v2f a2 = gp[2 * 32 + lane];
    v2f a3 = gp[3 * 32 + lane];

    v8f d0 = __builtin_amdgcn_wmma_f32_16x16x4_f32(false, a0, false, bm, (short)0, zero, false, false);
    v8f d1 = __builtin_amdgcn_wmma_f32_16x16x4_f32(false, a1, false, bm, (short)0, zero, false, false);
    v8f d2 = __builtin_amdgcn_wmma_f32_16x16x4_f32(false, a2, false, bm, (short)0, zero, false, false);
    v8f dz = __builtin_amdgcn_wmma_f32_16x16x4_f32(false, a3, false, bm, (short)0, zero, false, false);

#pragma unroll
    for (int j = 0; j < 8; ++j) {
      float w0 = d0[j], w1 = d1[j], w2 = d2[j], zp = dz[j];
      bool valid = (w0 >= 0.0f) && (w1 >= 0.0f) && (w2 >= 0.0f) && (zp > EPSZ);
      float z = valid ? zp : __builtin_inff();
      int fid = (g << 4) + j + (hi << 3);
      if (z < zmin) { zmin = z; idxm = fid; }
    }
  }

  // Combine the two lane halves (faces 0-7 vs 8-15 of each group).
  float zo = __shfl_xor(zmin, 16, 32);
  int io = __shfl_xor(idxm, 16, 32);
  if (zo < zmin || (zo == zmin && io < idxm)) { zmin = zo; idxm = io; }

  bool hit = (zmin < 1e30f);

  if (lane < 16) {
    const int gpix = b * H_ * W_ + row * W_ + col;
    float* o_p2f = out;
    float* o_z   = out + NP;
    float* o_by  = out + 2 * NP;  // (B,H,W,1,3)
    float* o_d   = out + 5 * NP;

    if (!hit) {
      o_p2f[gpix] = -1.0f;
      o_z[gpix] = -1.0f;
      o_by[3 * gpix + 0] = -1.0f;
      o_by[3 * gpix + 1] = -1.0f;
      o_by[3 * gpix + 2] = -1.0f;
      o_d[gpix] = -1.0f;
    } else {
      const float* fp = fv + ((size_t)b * F_ + idxm) * 9;
      float x0 = fp[0], y0 = fp[1], z0 = fp[2];
      float x1 = fp[3], y1 = fp[4], z1 = fp[5];
      float x2 = fp[6], y2 = fp[7], z2 = fp[8];

      float a2v = mr_edge(x0, y0, x1, y1, x2, y2);
      float a2s = (fabsf(a2v) > EPSA) ? a2v : EPSA;
      float c0 = mr_edge(x1, y1, x2, y2, px, py) / a2s;
      float c1 = mr_edge(x2, y2, x0, y0, px, py) / a2s;
      float c2 = mr_edge(x0, y0, x1, y1, px, py) / a2s;
      float zb = c0 * z0 + c1 * z1 + c2 * z2;

      float dd = fminf(fminf(mr_seg_d2(x0, y0, x1, y1, px, py),
                             mr_seg_d2(x1, y1, x2, y2, px, py)),
                       mr_seg_d2(x2, y2, x0, y0, px, py));
      bool ins = (c0 >= 0.0f) && (c1 >= 0.0f) && (c2 >= 0.0f);

      o_p2f[gpix] = (float)idxm;
      o_z[gpix] = zb;
      o_by[3 * gpix + 0] = c0;
      o_by[3 * gpix + 1] = c1;
      o_by[3 * gpix + 2] = c2;
      o_d[gpix] = ins ? -dd : dd;
    }
  }
}

// ---------------------------------------------------------------------------
extern "C" void kernel_launch(void* const* d_in, const int* in_sizes, int n_in,
                              void* d_out, int out_size, void* d_ws, size_t ws_size,
                              hipStream_t stream) {
  const float* verts = (const float*)d_in[0];
  const float* R     = (const float*)d_in[1];
  const float* T     = (const float*)d_in[2];
  const float* focal = (const float*)d_in[3];
  const int*   faces = (const int*)d_in[4];
  float* out = (float*)d_out;

  float* ws = (float*)d_ws;
  float* vs = ws;                                  // B*V*3   = 48000 floats
  float* fv = ws + (size_t)B_ * V_ * 3;            // B*F*9   = 27000 floats
  float2* apack = (float2*)(ws + (size_t)B_ * V_ * 3 + (size_t)B_ * F_ * 9);
  // apack: B*NGROUPS*4*32 float2 = 48128 floats; total ws ~ 481 KB

  {
    int total = B_ * V_;
    mr_project<<<(total + 255) / 256, 256, 0, stream>>>(verts, R, T, focal, vs);
  }
  {
    int total = B_ * FPAD;
    mr_facecoef<<<(total + 255) / 256, 256, 0, stream>>>(vs, faces, fv, apack);
  }
  {
    mr_raster<<<B_ * H_, 256, 0, stream>>>((const float*)apack, fv, out);
  }
}